// Scorpion_5626407158304
// MI455X (gfx1250) — compile-verified
//
#include <hip/hip_runtime.h>
#include <hip/hip_bf16.h>
#include <math.h>

// Problem constants (match reference): P=128 proxy sets, E=32 elems, D=64,
// B=1024 segments, N=131072.
#define P_ 128
#define E_ 32
#define D_ 64
#define B_ 1024

typedef __attribute__((ext_vector_type(2))) float v2f;
typedef __attribute__((ext_vector_type(8))) float v8f;

// ds_swizzle group-of-32 encodings: offset = (xor_mask<<10) | (or_mask<<5) | and_mask
#define SWZ_XOR1  0x041F
#define SWZ_XOR2  0x081F
#define SWZ_XOR4  0x101F
#define SWZ_XOR8  0x201F
#define SWZ_XOR16 0x401F

template <int IMM>
__device__ __forceinline__ float swz_f32(float v) {
    return __int_as_float(__builtin_amdgcn_ds_swizzle(__float_as_int(v), IMM));
}

// ---------------------------------------------------------------------------
// Kernel 0: exclusive scan of segment lengths -> segment start offsets.
// B_ = 1024 elements, single workgroup Hillis-Steele in LDS.
// ---------------------------------------------------------------------------
__global__ __launch_bounds__(1024) void moonpool_scan(const int* __restrict__ index,
                                                      int* __restrict__ offs) {
    __shared__ int s[B_];
    const int tid = threadIdx.x;
    s[tid] = index[tid];
    __syncthreads();
    for (int off = 1; off < B_; off <<= 1) {
        int v = 0;
        if (tid >= off) v = s[tid - off];
        __syncthreads();
        s[tid] += v;
        __syncthreads();
    }
    offs[tid] = (tid == 0) ? 0 : s[tid - 1];  // exclusive
}

// ---------------------------------------------------------------------------
// Kernel 1: main compute. One wave32 per (p, segment b).
//   sim(32x128) = proxy[p](32x64) * x[seg]^T(64x128) via V_WMMA_F32_16X16X4_F32
//   t[n] = max over 32 e-rows;  pooled[b][p] = sum_n t[n]
// Workgroup = 8 waves (256 threads), each wave owns one segment for this p.
// Branch-free inner loop: padded tail columns load a clamped (valid) address
// and their column-max is zeroed before accumulation.
// ---------------------------------------------------------------------------
__global__ __launch_bounds__(256) void moonpool_main(
    const float* __restrict__ x,          // [N, 64]
    const float* __restrict__ proxy,      // [128, 32, 64]
    const int*   __restrict__ index,      // [1024] segment lengths
    const int*   __restrict__ offs,       // [1024] segment start offsets
    float*       __restrict__ pooled)     // [1024, 128]
{
    const int p    = blockIdx.x;
    const int wave = threadIdx.x >> 5;
    const int lane = threadIdx.x & 31;
    const int b    = blockIdx.y * 8 + wave;

    // WMMA f32 16x16x4 A-fragment layout: lane m = L&15 holds row M=m;
    // lanes 0-15 carry K={0,1}, lanes 16-31 carry K={2,3}.
    const int m     = lane & 15;
    const int khalf = (lane >> 4) ? 2 : 0;

    // Preload both e-halves of proxy[p] for all 16 K-steps into registers
    // (2 halves x 16 steps x float2 = 64 VGPRs), reused across the segment.
    v2f a0[16], a1[16];
    const float* arow0 = proxy + ((size_t)p * E_ + m)      * D_ + khalf;  // e = m
    const float* arow1 = proxy + ((size_t)p * E_ + 16 + m) * D_ + khalf;  // e = 16+m
    #pragma unroll
    for (int k = 0; k < 16; ++k) {
        a0[k] = *(const v2f*)(arow0 + 4 * k);
        a1[k] = *(const v2f*)(arow1 + 4 * k);
    }

    const int seg_start = offs[b];
    const int seg_len   = index[b];
    const int ntiles    = (seg_len + 15) >> 4;

    // Per-lane accumulator: sum over tiles of this lane's column max.
    float acc = 0.f;

    for (int nt = 0; nt < ntiles; ++nt) {
        const int ncol   = (nt << 4) + m;                       // column within segment
        const int nclamp = (ncol < seg_len) ? ncol : (seg_len - 1);
        const float* xrow = x + (size_t)(seg_start + nclamp) * D_ + khalf;

        v8f c0 = {}; // e rows 0..15  x 16 cols
        v8f c1 = {}; // e rows 16..31 x 16 cols
        #pragma unroll
        for (int k = 0; k < 16; ++k) {
            const v2f bfrag = *(const v2f*)(xrow + 4 * k);      // unconditional, clamped
            c0 = __builtin_amdgcn_wmma_f32_16x16x4_f32(
                    false, a0[k], false, bfrag, (short)0, c0, false, false);
            c1 = __builtin_amdgcn_wmma_f32_16x16x4_f32(
                    false, a1[k], false, bfrag, (short)0, c1, false, false);
        }

        // Per-lane max over the 16 e-values this lane holds
        // (C layout: VGPR i -> M=i for lanes 0-15, M=i+8 for lanes 16-31).
        float t = fmaxf(c0[0], c0[1]);
        #pragma unroll
        for (int i = 2; i < 8; ++i) t = fmaxf(t, c0[i]);
        #pragma unroll
        for (int i = 0; i < 8; ++i) t = fmaxf(t, c1[i]);
        // Merge complementary 16 e-values held by lane^16 (same column):
        // single ds_swizzle SWAPX16, no address VGPR, one DS op per tile.
        t = fmaxf(t, swz_f32<SWZ_XOR16>(t));
        // Padded tail columns contribute nothing.
        if (ncol >= seg_len) t = 0.f;
        acc += t;
    }

    // One final cross-lane sum: lanes 0-15 and 16-31 hold identical values
    // per column slot, so a 16-lane butterfly inside each half yields the
    // full segment sum in every lane of that half.
    acc += swz_f32<SWZ_XOR1>(acc);
    acc += swz_f32<SWZ_XOR2>(acc);
    acc += swz_f32<SWZ_XOR4>(acc);
    acc += swz_f32<SWZ_XOR8>(acc);

    if (lane == 0) pooled[(size_t)b * P_ + p] = acc;
}

// ---------------------------------------------------------------------------
// Kernel 2: row-wise L2 normalize pooled[b, :] over P=128.
// One block per row; in-place is safe (block owns its row, barrier between
// read and write).
// ---------------------------------------------------------------------------
__global__ __launch_bounds__(128) void moonpool_norm(float* __restrict__ out) {
    const int b = blockIdx.x;
    const int p = threadIdx.x;
    float v = out[(size_t)b * P_ + p];
    float s = v * v;
    s += swz_f32<SWZ_XOR1>(s);
    s += swz_f32<SWZ_XOR2>(s);
    s += swz_f32<SWZ_XOR4>(s);
    s += swz_f32<SWZ_XOR8>(s);
    s += swz_f32<SWZ_XOR16>(s);
    __shared__ float wsum[4];
    const int wave = threadIdx.x >> 5;
    if ((threadIdx.x & 31) == 0) wsum[wave] = s;
    __syncthreads();
    const float tot  = wsum[0] + wsum[1] + wsum[2] + wsum[3];
    const float norm = fmaxf(sqrtf(tot), 1e-12f);
    out[(size_t)b * P_ + p] = v / norm;
}

// ---------------------------------------------------------------------------
extern "C" void kernel_launch(void* const* d_in, const int* in_sizes, int n_in,
                              void* d_out, int out_size, void* d_ws, size_t ws_size,
                              hipStream_t stream) {
    const float* x     = (const float*)d_in[0];   // [N, 64] fp32
    const float* proxy = (const float*)d_in[1];   // [128, 32, 64] fp32
    const int*   index = (const int*)d_in[2];     // [1024] int32
    float*       out   = (float*)d_out;           // [1024, 128] fp32
    int*         offs  = (int*)d_ws;              // 4 KB scratch for offsets

    (void)in_sizes; (void)n_in; (void)out_size; (void)ws_size;

    moonpool_scan<<<1, 1024, 0, stream>>>(index, offs);

    dim3 grid(P_, B_ / 8);
    moonpool_main<<<grid, 256, 0, stream>>>(x, proxy, index, offs, out);

    moonpool_norm<<<B_, 128, 0, stream>>>(out);
}